// LevelSetLayer_68813966017172
// MI455X (gfx1250) — compile-verified
//
#include <hip/hip_runtime.h>
#include <stdint.h>

// ---------------- problem constants (48x48 grid complex) ----------------
#define HH 48
#define WW 48
#define NV (HH*WW)                 // 2304 vertices
#define NEH (HH*(WW-1))            // 2256 horizontal edges
#define NEV ((HH-1)*WW)            // 2256 vertical edges
#define NED ((HH-1)*(WW-1))        // 2209 anti-diagonal edges
#define NE (NEH+NEV+NED)           // 6721 edges
#define NF (2*(HH-1)*(WW-1))       // 4418 triangles
#define M  (NV+NE+NF)              // 13443 simplices
#define MSORT 16384                // next pow2 for bitonic sort
#define WPC 432                    // u32 words per bit-packed column (>= ceil(M/32)=421, 16B-multiple)
#define RBS 512                    // reduction block size (16 wave32)

// ---------------- static device scratch (graph-capture safe) ------------
__device__ float    g_vals[M];
__device__ uint64_t g_keys[MSORT];
__device__ int      g_order[M];
__device__ int      g_rank[M];
__device__ int      g_pivot[M];
__device__ int      g_lowR[M];
__device__ float    g_fmax;
__device__ unsigned g_R[(size_t)M * WPC];   // ~22.2 MB, L2-resident on MI455X

__device__ __forceinline__ unsigned ordf(float f) {
  // order-preserving float -> uint32 map (ascending)
  unsigned u = __float_as_uint(f);
  return (u & 0x80000000u) ? ~u : (u | 0x80000000u);
}

// 1) filtration: vertices + edges (edge value = max of endpoint values)
__global__ void k_filt_ve(const float* __restrict__ img, const int* __restrict__ edges) {
  int i = blockIdx.x * blockDim.x + threadIdx.x;
  if (i < NV) {
    g_vals[i] = img[i];
  } else if (i < NV + NE) {
    int e = i - NV;
    float a = img[edges[2*e]];
    float b = img[edges[2*e+1]];
    g_vals[i] = fmaxf(a, b);
  }
}

// 2) filtration: triangles (max of edge maxes) + build 48-bit sort keys
//    key = ord(val)[32] : dim[2] : original index[14]  -> stable lexsort((dims, vals))
__global__ void k_filt_tri_keys(const int* __restrict__ tri) {
  int i = blockIdx.x * blockDim.x + threadIdx.x;
  if (i >= MSORT) return;
  if (i >= M) { g_keys[i] = ~0ull; return; }   // pad sorts to the end
  float v; int dim;
  if (i < NV)           { v = g_vals[i]; dim = 0; }
  else if (i < NV + NE) { v = g_vals[i]; dim = 1; }
  else {
    int t = i - NV - NE;
    float a = g_vals[NV + tri[3*t + 0]];
    float b = g_vals[NV + tri[3*t + 1]];
    float c = g_vals[NV + tri[3*t + 2]];
    v = fmaxf(a, fmaxf(b, c));
    g_vals[i] = v;
    dim = 2;
  }
  g_keys[i] = ((uint64_t)ordf(v) << 16) | ((uint64_t)dim << 14) | (uint64_t)i;
}

// 3) single-workgroup bitonic sort, fully LDS-resident (128 KB of the 320 KB WGP LDS)
__global__ __launch_bounds__(1024) void k_sort() {
  extern __shared__ uint64_t s[];
  const int tid = threadIdx.x;
  for (int i = tid; i < MSORT; i += 1024) s[i] = g_keys[i];
  __syncthreads();
  for (int k = 2; k <= MSORT; k <<= 1) {
    for (int jj = k >> 1; jj > 0; jj >>= 1) {
      for (int i = tid; i < MSORT; i += 1024) {
        int ixj = i ^ jj;
        if (ixj > i) {
          uint64_t a = s[i], b = s[ixj];
          bool up = ((i & k) == 0);
          if ((a > b) == up) { s[i] = b; s[ixj] = a; }
        }
      }
      __syncthreads();
    }
  }
  for (int p = tid; p < M; p += 1024) {
    int oi = (int)(s[p] & 0x3FFFull);
    g_order[p] = oi;
    g_rank[oi] = p;
  }
  if (tid == 0) g_fmax = g_vals[(int)(s[M-1] & 0x3FFFull)];
}

// 4) zero the bit-packed boundary matrix
__global__ void k_clearR() {
  size_t n = (size_t)M * WPC;
  size_t stride = (size_t)gridDim.x * blockDim.x;
  for (size_t i = (size_t)blockIdx.x * blockDim.x + threadIdx.x; i < n; i += stride)
    g_R[i] = 0u;
}

// 5) scatter boundary bits (each thread owns exactly one column -> no atomics needed)
__global__ void k_boundary(const int* __restrict__ edges, const int* __restrict__ tri) {
  int i = blockIdx.x * blockDim.x + threadIdx.x;
  if (i < NE) {
    int col = g_rank[NV + i];
    int r0  = g_rank[edges[2*i]];
    int r1  = g_rank[edges[2*i+1]];
    size_t base = (size_t)col * WPC;
    int w0 = r0 >> 5, w1 = r1 >> 5;
    unsigned b0 = 1u << (r0 & 31), b1 = 1u << (r1 & 31);
    if (w0 == w1) g_R[base + w0] = b0 | b1;
    else { g_R[base + w0] = b0; g_R[base + w1] = b1; }
  } else if (i < NE + NF) {
    int t = i - NE;
    int col = g_rank[NV + NE + t];
    size_t base = (size_t)col * WPC;
    int w[3]; unsigned b[3];
    for (int k = 0; k < 3; ++k) {
      int r = g_rank[NV + tri[3*t + k]];
      w[k] = r >> 5; b[k] = 1u << (r & 31);
    }
    for (int k = 1; k < 3; ++k)
      for (int l = 0; l < k; ++l)
        if (w[k] >= 0 && w[k] == w[l]) { b[l] |= b[k]; w[k] = -1; }
    for (int k = 0; k < 3; ++k)
      if (w[k] >= 0) g_R[base + w[k]] = b[k];
  }
}

// 6) persistence column reduction with the twist/clearing optimization.
//    Pass 0: dim-2 columns (triangles) in filtration order; each pivot row found
//            clears the corresponding edge column (twist lemma: it reduces to zero).
//    Pass 1: dim-1 columns (edges) that were not cleared; vertices are trivially zero.
//    Working column + pivot map live in LDS; pivot columns staged L2 -> LDS with
//    gfx1250 async-to-LDS b128 transfers.
__global__ __launch_bounds__(RBS) void k_reduce() {
  __shared__ __align__(16) unsigned s_cur[WPC];
  __shared__ __align__(16) unsigned s_pcol[WPC];
  __shared__ int s_pivarr[M];     // row -> column that has this row as its low
  __shared__ int s_low;
  const int tid = threadIdx.x;

  for (int i = tid; i < M; i += RBS) s_pivarr[i] = -1;
  unsigned* Rg = g_R;
  const uint64_t Rbase = (uint64_t)(size_t)Rg;
  __syncthreads();

  for (int pass = 0; pass < 2; ++pass) {
    for (int j = 0; j < M; ++j) {
      int oi  = g_order[j];                                  // uniform scalar
      int dim = (oi < NV) ? 0 : ((oi < NV + NE) ? 1 : 2);
      if (pass == 0) {
        if (dim != 2) continue;                              // triangles only
      } else {
        if (dim == 2) continue;                              // already done
        if (dim == 0) { if (tid == 0) g_lowR[j] = -1; continue; }   // empty boundary
        if (s_pivarr[j] >= 0) {                              // twist: cleared edge column
          if (tid == 0) g_lowR[j] = -1;
          continue;
        }
      }

      // prefetch the next column while working on this one (global_prefetch_b8)
      if (j + 1 < M && tid < 27)
        __builtin_prefetch(&g_R[(size_t)(j + 1) * WPC + (size_t)tid * 16], 0, 1);

      // stage column j into LDS (b128 per lane)
      if (tid < WPC / 4)
        ((uint4*)s_cur)[tid] = ((const uint4*)(Rg + (size_t)j * WPC))[tid];
      if (tid == 0) s_low = -1;
      __syncthreads();
      if (tid < WPC) {
        unsigned x = s_cur[tid];
        if (x) atomicMax(&s_low, (tid << 5) + (31 - __builtin_clz(x)));  // ds_max_rtn
      }
      __syncthreads();
      int l = s_low;

      while (l >= 0 && s_pivarr[l] >= 0) {
        int pv = s_pivarr[l];
        // async copy pivot column pv from global (L2-resident) directly into LDS
        if (tid < WPC / 4) {
          unsigned ldsA = (unsigned)(size_t)&s_pcol[tid * 4];   // low 32b of generic ptr = LDS offset
          unsigned off  = (unsigned)(((size_t)pv * WPC + (size_t)tid * 4) * 4u);
          asm volatile("global_load_async_to_lds_b128 %0, %1, %2"
                       :: "v"(ldsA), "v"(off), "s"(Rbase) : "memory");
        }
        asm volatile("s_wait_asynccnt 0" ::: "memory");
        __syncthreads();
        if (tid == 0) s_low = -1;
        __syncthreads();
        if (tid < WPC) {
          unsigned x = s_cur[tid] ^ s_pcol[tid];                // GF(2) column elimination
          s_cur[tid] = x;
          if (x) atomicMax(&s_low, (tid << 5) + (31 - __builtin_clz(x)));
        }
        __syncthreads();
        l = s_low;
      }

      if (tid == 0) {
        g_lowR[j] = l;
        if (l >= 0) s_pivarr[l] = j;
      }
      if (tid < WPC / 4)
        ((uint4*)(Rg + (size_t)j * WPC))[tid] = ((const uint4*)s_cur)[tid];
      __threadfence();   // reduced column must be visible before later use as a pivot
      __syncthreads();
    }
    __syncthreads();
  }

  for (int i = tid; i < M; i += RBS) g_pivot[i] = s_pivarr[i];
}

// 7) assemble diagrams: (2, M, 2) f32, zero elsewhere
__global__ void k_diag(float* __restrict__ out) {
  int i = blockIdx.x * blockDim.x + threadIdx.x;
  if (i >= M) return;
  int oi = g_order[i];
  float vs = g_vals[oi];
  int pv = g_pivot[i];
  bool paired = pv >= 0;
  bool essential = (g_lowR[i] < 0) && !paired;
  float dpair = paired ? g_vals[g_order[pv]] : 0.0f;
  float fm = g_fmax;
  int dim = (oi < NV) ? 0 : ((oi < NV + NE) ? 1 : 2);
  for (int d = 0; d < 2; ++d) {
    bool pd = paired && (dim == d);
    bool ed = essential && (dim == d);
    float birth = (pd || ed) ? vs : 0.0f;
    float death = pd ? dpair : (ed ? fm : 0.0f);
    out[((size_t)d * M + i) * 2 + 0] = birth;
    out[((size_t)d * M + i) * 2 + 1] = death;
  }
}

extern "C" void kernel_launch(void* const* d_in, const int* in_sizes, int n_in,
                              void* d_out, int out_size, void* d_ws, size_t ws_size,
                              hipStream_t stream) {
  const float* img   = (const float*)d_in[0];   // (48*48,) f32
  const int*   edges = (const int*)d_in[1];     // (E,2) i32 flat
  const int*   tri   = (const int*)d_in[2];     // (F,3) i32 flat
  float*       out   = (float*)d_out;           // (2, M, 2) f32 flat = 53772

  k_filt_ve      <<<(NV + NE + 255) / 256, 256, 0, stream>>>(img, edges);
  k_filt_tri_keys<<<MSORT / 256,           256, 0, stream>>>(tri);
  k_sort         <<<1, 1024, MSORT * sizeof(uint64_t), stream>>>();
  k_clearR       <<<2048, 256, 0, stream>>>();
  k_boundary     <<<(NE + NF + 255) / 256, 256, 0, stream>>>(edges, tri);
  k_reduce       <<<1, RBS, 0, stream>>>();
  k_diag         <<<(M + 255) / 256, 256, 0, stream>>>(out);
}